// SignatureModel_62886911148776
// MI455X (gfx1250) — compile-verified
//
#include <hip/hip_runtime.h>

// ---- problem constants (from reference) ----
constexpr int BATCH   = 64;
constexpr int SEQ     = 512;
constexpr int INCH    = 3;
constexpr int GROUPS_ = 2;
constexpr int EXTRA_  = 12;
constexpr int AUGCH   = 16;                 // 1 + 3 + 12
constexpr int HIDDEN_ = 256;
constexpr int OUTCH   = 32;
constexpr int SIG_PG  = AUGCH + AUGCH*AUGCH + AUGCH*AUGCH*AUGCH;  // 4368
constexpr int SIG_TOT = GROUPS_ * SIG_PG;   // 8736

typedef float v2f __attribute__((ext_vector_type(2)));
typedef float v8f __attribute__((ext_vector_type(8)));

// ============================================================
// Kernel A: depth-3 path signature via Chen's identity.
// One block per path (b*2+g). Thread t=(a*16+bq) owns S2[a,bq]
// and S3[a,bq,0..15] in registers. x[b] staged in LDS (stride-4).
// Level-3 update collapsed to 1 FMA/element:
//   S3[a,b,c] += (S2[a,b] + da*db/6 + 0.5*S1[a]*db) * d[c]
// ============================================================
__global__ __launch_bounds__(256) void sig_kernel(
    const float* __restrict__ x, const long long* __restrict__ lengths,
    const float* __restrict__ aug_w, float* __restrict__ sig_out)
{
  __shared__ float4 xl[SEQ];                      // 8 KB
  const int p   = blockIdx.x;                     // path = b*GROUPS + g
  const int bb  = p >> 1;
  const int g   = p & 1;
  const int tid = threadIdx.x;

  // cooperative stage of x[bb] (512x3) into LDS, padded to float4
  for (int r = tid; r < SEQ; r += 256) {
    const float* xr = x + ((size_t)bb * SEQ + r) * INCH;
    xl[r] = make_float4(xr[0], xr[1], xr[2], 0.f);
  }
  __syncthreads();

  // augmentation weights for this group (12x3) -> registers (uniform loads)
  float W[EXTRA_][INCH];
  #pragma unroll
  for (int e = 0; e < EXTRA_; ++e)
    #pragma unroll
    for (int i = 0; i < INCH; ++i)
      W[e][i] = aug_w[((size_t)g * EXTRA_ + e) * INCH + i];

  const int a  = tid >> 4;
  const int bq = tid & 15;
  const int lenm1 = (int)lengths[bb] - 1;
  const float DT = 1.0f / (float)(SEQ - 1);

  float S1a = 0.f, S2ab = 0.f;
  float S3[16];
  #pragma unroll
  for (int c = 0; c < 16; ++c) S3[c] = 0.f;

  float4 prev = xl[0];
  for (int s = 0; s < SEQ - 1; ++s) {
    float4 nxt = xl[s + 1];                       // LDS broadcast
    const float m = (s < lenm1) ? 1.f : 0.f;      // become_constant trick
    const float dx0 = (nxt.x - prev.x) * m;
    const float dx1 = (nxt.y - prev.y) * m;
    const float dx2 = (nxt.z - prev.z) * m;
    prev = nxt;

    float d[16];
    d[0] = DT; d[1] = dx0; d[2] = dx1; d[3] = dx2;
    #pragma unroll
    for (int e = 0; e < EXTRA_; ++e)              // bias cancels in increments
      d[4 + e] = fmaf(W[e][0], dx0, fmaf(W[e][1], dx1, W[e][2] * dx2));

    const float da = d[a], db = d[bq];
    // coefficient uses OLD S1/S2 (Chen carry)
    const float coef = fmaf(fmaf(da, (1.f/6.f), 0.5f * S1a), db, S2ab);
    #pragma unroll
    for (int c = 0; c < 16; ++c)
      S3[c] = fmaf(coef, d[c], S3[c]);
    S2ab = fmaf(fmaf(da, 0.5f, S1a), db, S2ab);   // S2 += (da/2 + S1old)*db
    S1a += da;                                    // S1 += d
  }

  float* sig = sig_out + (size_t)bb * SIG_TOT + (size_t)g * SIG_PG;
  if (bq == 0) sig[a] = S1a;                      // level 1 (16)
  sig[AUGCH + a * 16 + bq] = S2ab;                // level 2 (256)
  float* s3p = sig + AUGCH + AUGCH*AUGCH + (a * 16 + bq) * 16;
  #pragma unroll
  for (int c = 0; c < 16; ++c) s3p[c] = S3[c];    // level 3 (4096)
}

// ============================================================
// fp32 WMMA 16x16xK GEMM tile: D(16x16) += A(16x4) * B(4x16)
// A lanes 0-15: {K=k0,k0+1}; lanes 16-31: {K=k0+2,k0+3} (row M=l)
// B mirrored; C VGPR r -> rows (r, r+8), col = l.
// ============================================================
__device__ __forceinline__ v8f wmma_tile_f32(
    const float* __restrict__ A, int lda,   // row-major MxK, row m0 pre-applied
    const float* __restrict__ Bm, int ldb,  // row-major KxN, col n0+l pre-applied
    int K, int l, int half)
{
  v8f c = {};
  const float* arow = A + (size_t)l * lda;
  for (int k0 = 0; k0 < K; k0 += 4) {
    const int ka = k0 + 2 * half;
    v2f aa = *(const v2f*)(arow + ka);            // 8B-aligned (ka even)
    v2f bb;
    bb.x = Bm[(size_t)ka * ldb];
    bb.y = Bm[(size_t)(ka + 1) * ldb];
    c = __builtin_amdgcn_wmma_f32_16x16x4_f32(false, aa, false, bb,
                                              (short)0, c, false, false);
  }
  return c;
}

// Kernel B: h = relu(sig @ w1 + b1); M=64,N=256,K=8736; 64 tiles = 8 blk x 8 waves
__global__ __launch_bounds__(256) void mlp1_kernel(
    const float* __restrict__ sig, const float* __restrict__ w1,
    const float* __restrict__ b1, float* __restrict__ h)
{
  const int tid = threadIdx.x, wave = tid >> 5, lane = tid & 31;
  const int half = lane >> 4, l = lane & 15;
  const int tile = blockIdx.x * 8 + wave;         // 4 x 16 tiles
  const int m0 = (tile >> 4) * 16, n0 = (tile & 15) * 16;

  v8f c = wmma_tile_f32(sig + (size_t)m0 * SIG_TOT, SIG_TOT,
                        w1 + n0 + l, HIDDEN_, SIG_TOT, l, half);
  const float bias = b1[n0 + l];
  #pragma unroll
  for (int r = 0; r < 8; ++r) {
    const int row = m0 + r + 8 * half;
    const float v = c[r] + bias;
    h[(size_t)row * HIDDEN_ + n0 + l] = v > 0.f ? v : 0.f;
  }
}

// Kernel C: out = h @ w2 + b2; M=64,N=32,K=256; 8 tiles = 1 blk x 8 waves
__global__ __launch_bounds__(256) void mlp2_kernel(
    const float* __restrict__ h, const float* __restrict__ w2,
    const float* __restrict__ b2, float* __restrict__ out)
{
  const int tid = threadIdx.x, wave = tid >> 5, lane = tid & 31;
  const int half = lane >> 4, l = lane & 15;
  const int m0 = (wave >> 1) * 16, n0 = (wave & 1) * 16;

  v8f c = wmma_tile_f32(h + (size_t)m0 * HIDDEN_, HIDDEN_,
                        w2 + n0 + l, OUTCH, HIDDEN_, l, half);
  const float bias = b2[n0 + l];
  #pragma unroll
  for (int r = 0; r < 8; ++r) {
    const int row = m0 + r + 8 * half;
    out[(size_t)row * OUTCH + n0 + l] = c[r] + bias;
  }
}

extern "C" void kernel_launch(void* const* d_in, const int* in_sizes, int n_in,
                              void* d_out, int out_size, void* d_ws, size_t ws_size,
                              hipStream_t stream) {
  const float*     x       = (const float*)d_in[0];
  const long long* lengths = (const long long*)d_in[1];
  const float*     aug_w   = (const float*)d_in[2];
  // d_in[3] = aug_b: unused — signature depends only on increments, bias cancels
  const float*     w1      = (const float*)d_in[4];
  const float*     b1      = (const float*)d_in[5];
  const float*     w2      = (const float*)d_in[6];
  const float*     b2      = (const float*)d_in[7];

  float* sig = (float*)d_ws;                        // 64*8736 floats (2.24 MB)
  float* h   = sig + (size_t)BATCH * SIG_TOT;       // 64*256  floats

  sig_kernel<<<dim3(BATCH * GROUPS_), dim3(256), 0, stream>>>(x, lengths, aug_w, sig);
  mlp1_kernel<<<dim3(8), dim3(256), 0, stream>>>(sig, w1, b1, h);
  mlp2_kernel<<<dim3(1), dim3(256), 0, stream>>>(h, w2, b2, (float*)d_out);
}